// SDNEncoder_65970697666823
// MI455X (gfx1250) — compile-verified
//
#include <hip/hip_runtime.h>
#include <math.h>

// ---------------- constants from the reference ----------------
constexpr int Bsz  = 256;
constexpr int NN   = 64;     // nodes
constexpr int NG   = 16;     // groups
constexpr int L    = 64;
constexpr int CC   = 32;     // conv channels
constexpr int GH   = 128;
constexpr int OUTD = 64;
constexpr int E1   = 262144;
constexpr int E2   = 32768;
constexpr int FEAT = 6 * CC; // 192

typedef __attribute__((ext_vector_type(2))) float v2f;
typedef __attribute__((ext_vector_type(8))) float v8f;

// Hardware float atomic add (GLOBAL_ATOMIC_ADD_F32) — avoid CAS loops.
__device__ __forceinline__ void atomic_fadd(float* p, float v) {
  unsafeAtomicAdd(p, v);
}

// =====================================================================
// Generic "expert" GEMM using V_WMMA_F32_16X16X4_F32.
//   out[m, e, :] = relu?( A_row(m,e) @ W[e] + bias[e] )
//   A_row(m,e) = A + m*strideA + e*offA           (K floats)
//   W[e]       = W + e*K*Nd                        (K x Nd row-major)
//   out        = out + m*strideOut + e*offOut      (Nd floats)
// One wave computes one 16x16 tile of C.
// A-layout (ISA 7.12.2, 32-bit A 16x4): lane m = lane&15, VGPR0/1 hold
// K = (lane>>4)*2 + {0,1}. B 4x16: VGPRj low half row K=j, high half K=j+2.
// C 16x16: VGPRj low half M=j, high half M=j+8; N = lane&15.
// =====================================================================
__global__ void expert_gemm_f32_wmma(
    const float* __restrict__ A, const float* __restrict__ W,
    const float* __restrict__ bias, float* __restrict__ out,
    int totalWaves, int ntiles, int tilesPerExpert,
    int K, int Nd,
    int strideA, int offA, int strideOut, int offOut, int relu)
{
  int gtid = blockIdx.x * blockDim.x + threadIdx.x;
  int wave = gtid >> 5;
  if (wave >= totalWaves) return;           // wave-uniform: EXEC stays all-1s
  int lane = threadIdx.x & 31;
  int e  = wave / tilesPerExpert;
  int t  = wave % tilesPerExpert;
  int mt = t / ntiles;
  int nt = t % ntiles;
  int m0 = mt * 16, n0 = nt * 16;
  int laneLow = lane & 15;
  int laneHi  = lane >> 4;                  // 0 or 1

  const float* __restrict__ Arow =
      A + (size_t)(m0 + laneLow) * strideA + (size_t)e * offA + laneHi * 2;
  const float* __restrict__ We =
      W + (size_t)e * K * Nd + n0 + laneLow;

  v8f acc = {};
  for (int k0 = 0; k0 < K; k0 += 4) {
    v2f a, b;
    a.x = Arow[k0];
    a.y = Arow[k0 + 1];
    int kB = k0 + laneHi * 2;
    b.x = We[(size_t)kB       * Nd];
    b.y = We[(size_t)(kB + 1) * Nd];
    acc = __builtin_amdgcn_wmma_f32_16x16x4_f32(
        false, a, false, b, (short)0, acc, false, false);
  }

  float bv = bias ? bias[(size_t)e * Nd + n0 + laneLow] : 0.0f;
  float* __restrict__ orow = out + (size_t)e * offOut + n0 + laneLow;
#pragma unroll
  for (int j = 0; j < 8; ++j) {
    int m = m0 + j + laneHi * 8;
    float v = acc[j] + bv;
    if (relu) v = fmaxf(v, 0.0f);
    orow[(size_t)m * strideOut] = v;
  }
}

// =====================================================================
// Per-node grouped conv (6 variants) + bias + time-max + ReLU -> feats
// One block per (b,n); 192 threads = 6 conv types x 32 channels.
// =====================================================================
__global__ void conv_feats_kernel(
    const float* __restrict__ x,
    const float* __restrict__ w3,  const float* __restrict__ b3,
    const float* __restrict__ w5,  const float* __restrict__ b5,
    const float* __restrict__ w7,  const float* __restrict__ b7,
    const float* __restrict__ tw3, const float* __restrict__ tb3,
    const float* __restrict__ tw5, const float* __restrict__ tb5,
    const float* __restrict__ tw7, const float* __restrict__ tb7,
    float* __restrict__ feats)
{
  int bn = blockIdx.x;            // b*NN + n
  int n  = bn & (NN - 1);
  __shared__ float xs[4 * L];
  const float* xrow = x + (size_t)bn * 4 * L;
  for (int i = threadIdx.x; i < 4 * L; i += blockDim.x) xs[i] = xrow[i];
  __syncthreads();

  int tid  = threadIdx.x;         // 0..191
  int type = tid >> 5;
  int ch   = tid & 31;

  const float* W; const float* Bv; int srcch, Lin, k, pad;
  switch (type) {
    case 0:  W = w3;  Bv = b3;  srcch = 0; Lin = 64; k = 3; pad = 1; break;
    case 1:  W = w5;  Bv = b5;  srcch = 0; Lin = 64; k = 5; pad = 2; break;
    case 2:  W = w7;  Bv = b7;  srcch = 0; Lin = 64; k = 7; pad = 3; break;
    case 3:  W = tw3; Bv = tb3; srcch = 1; Lin = 62; k = 3; pad = 1; break;
    case 4:  W = tw5; Bv = tb5; srcch = 2; Lin = 60; k = 5; pad = 2; break;
    default: W = tw7; Bv = tb7; srcch = 3; Lin = 58; k = 7; pad = 3; break;
  }
  const float* wp = W + ((size_t)n * CC + ch) * k;
  float wreg[7];
  for (int j = 0; j < k; ++j) wreg[j] = wp[j];

  const float* xin = xs + srcch * L;
  float mx = -INFINITY;
  for (int tt = 0; tt < Lin; ++tt) {    // 'same' padding: Lout == Lin
    float s = 0.0f;
    for (int j = 0; j < k; ++j) {
      int idx = tt - pad + j;
      float xv = (idx >= 0 && idx < Lin) ? xin[idx] : 0.0f;
      s = fmaf(xv, wreg[j], s);
    }
    mx = fmaxf(mx, s);
  }
  mx += Bv[(size_t)n * CC + ch];
  feats[(size_t)bn * FEAT + type * CC + ch] = fmaxf(mx, 0.0f);
}

// =====================================================================
// GCN helpers
// =====================================================================
__global__ void fill_f32(float* p, float v, int n) {
  int i = blockIdx.x * blockDim.x + threadIdx.x;
  if (i < n) p[i] = v;
}
__global__ void count_dst_kernel(const int* __restrict__ dst, int E, float* deg) {
  int i = blockIdx.x * blockDim.x + threadIdx.x;
  if (i < E) atomic_fadd(&deg[dst[i]], 1.0f);
}
__global__ void rsqrt_inplace(float* p, int n) {
  int i = blockIdx.x * blockDim.x + threadIdx.x;
  if (i < n) p[i] = rsqrtf(p[i]);
}
// out[i,c] = bias[c] + dinv[i]^2 * hw[i,c]   (self-loop message + bias)
__global__ void gcn_self_kernel(const float* __restrict__ hw,
                                const float* __restrict__ dinv,
                                const float* __restrict__ bias,
                                float* __restrict__ out, int nodes) {
  int i = blockIdx.x * blockDim.x + threadIdx.x;
  if (i >= nodes * GH) return;
  int node = i >> 7, c = i & (GH - 1);
  float di = dinv[node];
  out[i] = bias[c] + di * di * hw[i];
}
// one wave per edge: out[dst] += dinv[src]*dinv[dst] * hw[src]  (128 ch)
// per-lane: one global_load_b128 + 4 global_atomic_add_f32
__global__ void gcn_edge_kernel(const int* __restrict__ src,
                                const int* __restrict__ dst, int E,
                                const float* __restrict__ hw,
                                const float* __restrict__ dinv,
                                float* __restrict__ out) {
  int gtid = blockIdx.x * blockDim.x + threadIdx.x;
  int e = gtid >> 5;
  if (e >= E) return;
  int lane = threadIdx.x & 31;
  int s = src[e], d = dst[e];
  float coeff = dinv[s] * dinv[d];
  const float4* __restrict__ hs =
      (const float4*)(hw + (size_t)s * GH) + lane;
  float4 m = *hs;
  float* __restrict__ od = out + (size_t)d * GH + lane * 4;
  atomic_fadd(&od[0], coeff * m.x);
  atomic_fadd(&od[1], coeff * m.y);
  atomic_fadd(&od[2], coeff * m.z);
  atomic_fadd(&od[3], coeff * m.w);
}
__global__ void relu_inplace(float* p, int n) {
  int i = blockIdx.x * blockDim.x + threadIdx.x;
  if (i < n) p[i] = fmaxf(p[i], 0.0f);
}
// dcat[r, 0:128] = h1[r,:]; dcat[r,128:256] = h2[r,:]
__global__ void concat_kernel(const float* __restrict__ h1,
                              const float* __restrict__ h2,
                              float* __restrict__ dcat, int rows) {
  int i = blockIdx.x * blockDim.x + threadIdx.x;
  if (i >= rows * 256) return;
  int r = i >> 8, c = i & 255;
  dcat[i] = (c < 128) ? h1[(size_t)r * 128 + c] : h2[(size_t)r * 128 + (c - 128)];
}

// =====================================================================
// host side
// =====================================================================
static inline int cdiv(int a, int b) { return (a + b - 1) / b; }

extern "C" void kernel_launch(void* const* d_in, const int* in_sizes, int n_in,
                              void* d_out, int out_size, void* d_ws, size_t ws_size,
                              hipStream_t stream) {
  (void)in_sizes; (void)n_in; (void)out_size; (void)ws_size;

  const float* x1   = (const float*)d_in[0];
  const float* x2   = (const float*)d_in[1];
  const int* ei1    = (const int*)d_in[2];
  const int* ei2    = (const int*)d_in[3];
  const int* gei1   = (const int*)d_in[4];
  const int* gei2   = (const int*)d_in[5];
  const float* conv_w3 = (const float*)d_in[6];
  const float* conv_b3 = (const float*)d_in[7];
  const float* conv_w5 = (const float*)d_in[8];
  const float* conv_b5 = (const float*)d_in[9];
  const float* conv_w7 = (const float*)d_in[10];
  const float* conv_b7 = (const float*)d_in[11];
  const float* tconv_w3 = (const float*)d_in[12];
  const float* tconv_b3 = (const float*)d_in[13];
  const float* tconv_w5 = (const float*)d_in[14];
  const float* tconv_b5 = (const float*)d_in[15];
  const float* tconv_w7 = (const float*)d_in[16];
  const float* tconv_b7 = (const float*)d_in[17];
  const float* ff_w    = (const float*)d_in[18];
  const float* ff_b    = (const float*)d_in[19];
  const float* lower_w = (const float*)d_in[20];
  const float* lower_b = (const float*)d_in[21];
  const float* upper_w = (const float*)d_in[22];
  const float* upper_b = (const float*)d_in[23];
  const float* agg_w   = (const float*)d_in[24];
  const float* agg_b   = (const float*)d_in[25];
  const float* disc_w  = (const float*)d_in[26];
  const float* disc_b  = (const float*)d_in[27];
  const float* fc_w    = (const float*)d_in[28];
  const float* fc_b    = (const float*)d_in[29];

  // ---- workspace layout (floats) ----
  float* ws = (float*)d_ws;
  size_t o = 0;
  float* feats = ws + o; o += (size_t)Bsz * NN * FEAT;   // 3,145,728
  float* hnode = ws + o; o += (size_t)Bsz * NN * GH;     // 2,097,152
  float* hw    = ws + o; o += (size_t)Bsz * NN * GH;     // 2,097,152
  float* hgcn  = ws + o; o += (size_t)Bsz * NN * GH;     // 2,097,152
  float* hg    = ws + o; o += (size_t)Bsz * NG * GH;     //   524,288
  float* hgw   = ws + o; o += (size_t)Bsz * NG * GH;     //   524,288
  float* h1f   = ws + o; o += (size_t)Bsz * NG * GH;
  float* h2f   = ws + o; o += (size_t)Bsz * NG * GH;
  float* deg1  = ws + o; o += (size_t)Bsz * NN;
  float* deg2  = ws + o; o += (size_t)Bsz * NG;
  // dcat/ddisc alias the feats region (feats is dead after both encodes)
  float* dcat  = feats;                                  // 1,048,576
  float* ddisc = feats + (size_t)Bsz * NG * 256;         //   524,288

  auto gemm = [&](const float* A, const float* W, const float* bias, float* out,
                  int E, int M, int Nd, int K,
                  int sA, int oA, int sO, int oO, int relu) {
    int mtiles = M / 16, ntiles = Nd / 16;
    int tpe = mtiles * ntiles;
    int waves = E * tpe;
    int blocks = cdiv(waves * 32, 128);
    expert_gemm_f32_wmma<<<blocks, 128, 0, stream>>>(
        A, W, bias, out, waves, ntiles, tpe, K, Nd, sA, oA, sO, oO, relu);
  };

  auto encode = [&](const float* x, const int* ei, const int* gei, float* hf) {
    // conv + max + relu -> feats [B, NN, 192]
    conv_feats_kernel<<<Bsz * NN, 192, 0, stream>>>(
        x, conv_w3, conv_b3, conv_w5, conv_b5, conv_w7, conv_b7,
        tconv_w3, tconv_b3, tconv_w5, tconv_b5, tconv_w7, tconv_b7, feats);
    // per-node FF: [B,192] x [192,128] per node, + bias, relu -> hnode
    gemm(feats, ff_w, ff_b, hnode, NN, Bsz, GH, FEAT,
         NN * FEAT, FEAT, NN * GH, GH, 1);
    // node GCN: hw = hnode @ lower_w
    gemm(hnode, lower_w, nullptr, hw, 1, Bsz * NN, GH, GH, GH, 0, GH, 0, 0);
    // degrees (self-loop counts as 1) and dinv
    fill_f32<<<cdiv(Bsz * NN, 256), 256, 0, stream>>>(deg1, 1.0f, Bsz * NN);
    count_dst_kernel<<<cdiv(E1, 256), 256, 0, stream>>>(ei + E1, E1, deg1);
    rsqrt_inplace<<<cdiv(Bsz * NN, 256), 256, 0, stream>>>(deg1, Bsz * NN);
    // self term + bias, then edge scatter
    gcn_self_kernel<<<cdiv(Bsz * NN * GH, 256), 256, 0, stream>>>(
        hw, deg1, lower_b, hgcn, Bsz * NN);
    gcn_edge_kernel<<<cdiv(E1 * 32, 256), 256, 0, stream>>>(
        ei, ei + E1, E1, hw, deg1, hgcn);
    // per-group aggregate: rows (b,g) read 4*128 contiguous node features
    gemm(hgcn, agg_w, agg_b, hg, NG, Bsz, GH, 4 * GH,
         NN * GH, 4 * GH, NG * GH, GH, 0);
    // group GCN
    gemm(hg, upper_w, nullptr, hgw, 1, Bsz * NG, GH, GH, GH, 0, GH, 0, 0);
    fill_f32<<<cdiv(Bsz * NG, 256), 256, 0, stream>>>(deg2, 1.0f, Bsz * NG);
    count_dst_kernel<<<cdiv(E2, 256), 256, 0, stream>>>(gei + E2, E2, deg2);
    rsqrt_inplace<<<cdiv(Bsz * NG, 256), 256, 0, stream>>>(deg2, Bsz * NG);
    gcn_self_kernel<<<cdiv(Bsz * NG * GH, 256), 256, 0, stream>>>(
        hgw, deg2, upper_b, hf, Bsz * NG);
    gcn_edge_kernel<<<cdiv(E2 * 32, 256), 256, 0, stream>>>(
        gei, gei + E2, E2, hgw, deg2, hf);
    relu_inplace<<<cdiv(Bsz * NG * GH, 256), 256, 0, stream>>>(hf, Bsz * NG * GH);
  };

  encode(x1, ei1, gei1, h1f);
  encode(x2, ei2, gei2, h2f);

  // concat [B,16,256]
  concat_kernel<<<cdiv(Bsz * NG * 256, 256), 256, 0, stream>>>(h1f, h2f, dcat, Bsz * NG);
  // per-group disc: [B,256]x[256,128] + bias, relu -> ddisc [B, 16*128]
  gemm(dcat, disc_w, disc_b, ddisc, NG, Bsz, GH, 2 * GH,
       NG * 2 * GH, 2 * GH, NG * GH, GH, 1);
  // final fc: [256,2048] x [2048,64] + bias -> d_out
  gemm(ddisc, fc_w, fc_b, (float*)d_out, 1, Bsz, OUTD, NG * GH,
       NG * GH, 0, OUTD, 0, 0);
}